// GCN_35845797052945
// MI455X (gfx1250) — compile-verified
//
#include <hip/hip_runtime.h>
#include <hip/hip_bf16.h>

typedef float v2f __attribute__((ext_vector_type(2)));
typedef float v8f __attribute__((ext_vector_type(8)));

#define IN_CH  64
#define HID_CH 64
#define OUT_CH 32

// ---------------------------------------------------------------- degrees
__global__ __launch_bounds__(256) void k_init_deg(float* __restrict__ deg, int n) {
    int i = blockIdx.x * 256 + threadIdx.x;
    if (i < n) deg[i] = 1.0f;                 // self-loop contributes 1
}

__global__ __launch_bounds__(256) void k_deg(const int* __restrict__ dst,
                                             float* __restrict__ deg, int e) {
    int i = blockIdx.x * 256 + threadIdx.x;
    if (i < e) atomicAdd(&deg[dst[i]], 1.0f);
}

__global__ __launch_bounds__(256) void k_dis(const float* __restrict__ deg,
                                             float* __restrict__ dis, int n) {
    int i = blockIdx.x * 256 + threadIdx.x;
    if (i < n) dis[i] = rsqrtf(deg[i]);       // deg >= 1 always (self-loops)
}

// ---------------------------------------------------------------- WMMA GEMM
// C[M x NOUT] = A[M x 64] * B[64 x NOUT], fp32 via V_WMMA_F32_16X16X4_F32.
// One 16x16 tile per wave32; 8 waves per 256-thread block.
// A fragment (16x4): lanes 0-15 hold M=lane, K pair {0,1}; lanes 16-31 K pair {2,3}.
// B fragment (4x16): lanes hold N=lane&15, same K-pair split (mirror of A).
// C/D (16x16, 8 VGPRs): VGPR v -> M = v + (lane>>4)*8, N = lane&15.
template <int NOUT>
__global__ __launch_bounds__(256) void k_gemm_wmma(const float* __restrict__ A,
                                                   const float* __restrict__ B,
                                                   float* __restrict__ C,
                                                   int numTiles) {
    constexpr int K = 64;
    const int wave = threadIdx.x >> 5;
    const int lane = threadIdx.x & 31;
    const int tile = blockIdx.x * 8 + wave;
    if (tile >= numTiles) return;             // wave-uniform: EXEC stays all-1s
    constexpr int tilesPerRow = NOUT / 16;
    const int row0 = (tile / tilesPerRow) * 16;
    const int col0 = (tile % tilesPerRow) * 16;
    const int half = lane >> 4;               // 0: K={0,1}, 1: K={2,3}
    const int l15  = lane & 15;

    const float* __restrict__ Arow = A + (size_t)(row0 + l15) * K + half * 2;
    const float* __restrict__ Bp   = B + col0 + l15;

    v8f acc = {0.f, 0.f, 0.f, 0.f, 0.f, 0.f, 0.f, 0.f};
#pragma unroll
    for (int k0 = 0; k0 < K; k0 += 4) {
        v2f a;
        a.x = Arow[k0];
        a.y = Arow[k0 + 1];
        v2f b;
        b.x = Bp[(size_t)(k0 + half * 2)     * NOUT];
        b.y = Bp[(size_t)(k0 + half * 2 + 1) * NOUT];
        acc = __builtin_amdgcn_wmma_f32_16x16x4_f32(
            /*neg_a=*/false, a, /*neg_b=*/false, b,
            /*c_mod=*/(short)0, acc, /*reuse_a=*/false, /*reuse_b=*/false);
    }

    float* Cp = C + (size_t)row0 * NOUT + col0 + l15;
#pragma unroll
    for (int v = 0; v < 8; ++v)
        Cp[(size_t)(v + half * 8) * NOUT] = acc[v];
}

// ---------------------------------------------------------------- edge scatter
// agg[dst] += h[src] * dis[src]*dis[dst]; 4 channels per thread (float4 gather).
template <int C>
__global__ __launch_bounds__(256) void k_scatter(const int* __restrict__ src,
                                                 const int* __restrict__ dst,
                                                 const float* __restrict__ h,
                                                 const float* __restrict__ dis,
                                                 float* __restrict__ agg, int e) {
    constexpr int TPE = C / 4;                // threads per edge
    int t = blockIdx.x * 256 + threadIdx.x;
    int ed = t / TPE;
    if (ed >= e) return;
    int c = (t % TPE) * 4;
    int s = src[ed], d = dst[ed];
    float norm = dis[s] * dis[d];
    const float4 v = *(const float4*)(h + (size_t)s * C + c);
    float* o = agg + (size_t)d * C + c;
    atomicAdd(o + 0, v.x * norm);
    atomicAdd(o + 1, v.y * norm);
    atomicAdd(o + 2, v.z * norm);
    atomicAdd(o + 3, v.w * norm);
}

// ---------------------------------------------------------------- layer-1 epilogue
// agg = relu(agg + h*dis^2 (self-loop) + bias)
__global__ __launch_bounds__(256) void k_finalize_relu(float* __restrict__ agg,
                                                       const float* __restrict__ h,
                                                       const float* __restrict__ dis,
                                                       const float* __restrict__ bias,
                                                       int n) {
    int t = blockIdx.x * 256 + threadIdx.x;
    int i = t / HID_CH;
    if (i >= n) return;
    int c = t % HID_CH;                       // t == i*HID_CH + c
    float di = dis[i];
    float v = agg[t] + h[t] * di * di + bias[c];
    agg[t] = fmaxf(v, 0.0f);
}

// ---------------------------------------------------------------- layer-2 epilogue + log_softmax
// wave32 == 32 output channels: one node per wave, lane == channel.
__global__ __launch_bounds__(256) void k_final_logsoftmax(const float* __restrict__ agg,
                                                          const float* __restrict__ h,
                                                          const float* __restrict__ dis,
                                                          const float* __restrict__ bias,
                                                          float* __restrict__ out, int n) {
    int wave = threadIdx.x >> 5;
    int lane = threadIdx.x & 31;
    int i = blockIdx.x * 8 + wave;
    if (i >= n) return;                       // wave-uniform
    float di = dis[i];
    size_t idx = (size_t)i * OUT_CH + lane;
    float v = agg[idx] + h[idx] * di * di + bias[lane];
    float m = v;
#pragma unroll
    for (int o = 16; o > 0; o >>= 1) m = fmaxf(m, __shfl_xor(m, o, 32));
    float s = __expf(v - m);
#pragma unroll
    for (int o = 16; o > 0; o >>= 1) s += __shfl_xor(s, o, 32);
    out[idx] = (v - m) - logf(s);
}

// ---------------------------------------------------------------- host launcher
extern "C" void kernel_launch(void* const* d_in, const int* in_sizes, int n_in,
                              void* d_out, int out_size, void* d_ws, size_t ws_size,
                              hipStream_t stream) {
    const float* x  = (const float*)d_in[0];
    const int*   ei = (const int*)  d_in[1];
    const float* W1 = (const float*)d_in[2];
    const float* b1 = (const float*)d_in[3];
    const float* W2 = (const float*)d_in[4];
    const float* b2 = (const float*)d_in[5];
    float* out = (float*)d_out;

    const int N = in_sizes[0] / IN_CH;        // 100000 (multiple of 16)
    const int E = in_sizes[1] / 2;            // 800000
    const int* src = ei;                      // edge_index[0]
    const int* dst = ei + E;                  // edge_index[1]

    // carve workspace
    char* ws = (char*)d_ws;
    size_t off = 0;
    auto carve = [&](size_t bytes) -> void* {
        void* p = ws + off;
        off += (bytes + 255) & ~(size_t)255;
        return p;
    };
    float* deg  = (float*)carve((size_t)N * sizeof(float));
    float* dis  = (float*)carve((size_t)N * sizeof(float));
    float* h1   = (float*)carve((size_t)N * HID_CH * sizeof(float));
    float* agg1 = (float*)carve((size_t)N * HID_CH * sizeof(float));
    float* h2   = (float*)carve((size_t)N * OUT_CH * sizeof(float));
    float* agg2 = (float*)carve((size_t)N * OUT_CH * sizeof(float));

    // degrees + normalization
    k_init_deg<<<(N + 255) / 256, 256, 0, stream>>>(deg, N);
    k_deg<<<(E + 255) / 256, 256, 0, stream>>>(dst, deg, E);
    k_dis<<<(N + 255) / 256, 256, 0, stream>>>(deg, dis, N);

    // ---- layer 1
    hipMemsetAsync(agg1, 0, (size_t)N * HID_CH * sizeof(float), stream);
    const int tiles1 = (N / 16) * (HID_CH / 16);
    k_gemm_wmma<HID_CH><<<(tiles1 + 7) / 8, 256, 0, stream>>>(x, W1, h1, tiles1);
    {
        long long threads = (long long)E * (HID_CH / 4);
        k_scatter<HID_CH><<<(unsigned)((threads + 255) / 256), 256, 0, stream>>>(
            src, dst, h1, dis, agg1, E);
    }
    {
        long long threads = (long long)N * HID_CH;
        k_finalize_relu<<<(unsigned)((threads + 255) / 256), 256, 0, stream>>>(
            agg1, h1, dis, b1, N);
    }

    // ---- layer 2
    hipMemsetAsync(agg2, 0, (size_t)N * OUT_CH * sizeof(float), stream);
    const int tiles2 = (N / 16) * (OUT_CH / 16);
    k_gemm_wmma<OUT_CH><<<(tiles2 + 7) / 8, 256, 0, stream>>>(agg1, W2, h2, tiles2);
    {
        long long threads = (long long)E * (OUT_CH / 4);
        k_scatter<OUT_CH><<<(unsigned)((threads + 255) / 256), 256, 0, stream>>>(
            src, dst, h2, dis, agg2, E);
    }
    k_final_logsoftmax<<<(N + 7) / 8, 256, 0, stream>>>(agg2, h2, dis, b2, out, N);
}